// UM_loss_26104811225770
// MI455X (gfx1250) — compile-verified
//
#include <hip/hip_runtime.h>
#include <hip/hip_bf16.h>

typedef __attribute__((ext_vector_type(2))) float v2f;
typedef __attribute__((ext_vector_type(8))) float v8f;

#define B_  64
#define T_  2000
#define C_  100
#define K_  200
#define D_  2048
#define TCHUNK 8
#define TPER (T_ / TCHUNK)          // 250
#define NBT 128                     // 64 act + 64 bkg
#define CHUNKS_PER_BT 16            // 2048 / (8 waves * 16 cols)
#define FEAT_BLOCKS (NBT * CHUNKS_PER_BT)   // 2048
#define SUP_BLOCKS  (B_ * TCHUNK)           // 512
#define SUP_SLOTS   (TCHUNK * 2)            // 16 partial sets (even/odd halves)

// ---------------------------------------------------------------------------
// Fused kernel: blocks [0,2048) do the f32-WMMA K-reduction for loss_um;
// blocks [2048,2560) do the per-(b,c) T-chunk partials for loss_sup.
// Both are independent, memory-bound streams; fusing removes a launch
// boundary and overlaps their tails.
// ---------------------------------------------------------------------------
__global__ void fused_partials(const float* __restrict__ feat_act,
                               const float* __restrict__ feat_bkg,
                               const float* __restrict__ gt,
                               const float* __restrict__ cas,
                               float* __restrict__ feat_part /*2048*/,
                               float* __restrict__ ssum_out /*[16][6400]*/,
                               float* __restrict__ pcnt_out /*[16][6400]*/) {
    if (blockIdx.x < FEAT_BLOCKS) {
        // ---------------- feat reduction via V_WMMA_F32_16X16X4_F32 --------
        // D = A(16x4 ones) x B(4x16 data) + C : every output column n
        // accumulates 4 K-rows of feat[b, :, d0+n] per instruction.
        const int bt    = blockIdx.x >> 4;      // 0..127 (act:0-63, bkg:64-127)
        const int chunk = blockIdx.x & 15;      // 0..15
        const int b     = bt & 63;
        const float* __restrict__ feat = (bt < B_) ? feat_act : feat_bkg;

        const int wave = threadIdx.x >> 5;
        const int lane = threadIdx.x & 31;
        const int col  = chunk * 128 + wave * 16 + (lane & 15);
        const int rowo = (lane >= 16) ? 2 : 0;  // lanes 16-31 carry K rows 2,3

        const size_t base = (size_t)b * K_ * D_ + (size_t)col;

        v2f ones; ones.x = 1.0f; ones.y = 1.0f;
        v8f acc = {};

        for (int k = 0; k < K_; k += 4) {
            const float* p = feat + base + (size_t)(k + rowo) * D_;
            v2f bm;
            bm.x = p[0];      // K row k+rowo
            bm.y = p[D_];     // K row k+rowo+1
            acc = __builtin_amdgcn_wmma_f32_16x16x4_f32(
                false, ones, false, bm, (short)0, acc, false, false);
        }

        // acc[0] lanes 0-15 hold column sums (lanes 16-31 duplicate row M=8)
        float mean = acc[0] * (1.0f / (float)K_);
        float msq  = (lane < 16) ? mean * mean : 0.0f;
        for (int off = 16; off > 0; off >>= 1)
            msq += __shfl_down(msq, off, 32);

        __shared__ float wsum[8];
        if (lane == 0) wsum[wave] = msq;
        __syncthreads();
        if (threadIdx.x == 0) {
            float s = 0.0f;
            #pragma unroll
            for (int i = 0; i < 8; ++i) s += wsum[i];
            feat_part[blockIdx.x] = s;   // partial of ||mean||^2 for this bt
        }
    } else {
        // ---------------- sup partials: (cas - (gt>0.5))^2 and pmask count --
        const int sblk  = blockIdx.x - FEAT_BLOCKS;  // 0..511
        const int b     = sblk & 63;
        const int chunk = sblk >> 6;                 // 0..7
        const int half  = threadIdx.x >> 7;          // 0 = even rows, 1 = odd
        const int c     = threadIdx.x & 127;
        if (c >= C_) return;

        const size_t base = (size_t)b * T_ * C_ + (size_t)c;
        const int tend = chunk * TPER + TPER;

        float ss = 0.0f, pc = 0.0f;
        for (int t = chunk * TPER + half; t < tend; t += 2) {
            const size_t idx = base + (size_t)t * C_;
            float g  = gt[idx];
            float cv = cas[idx];
            float pm = (g > 0.5f) ? 1.0f : 0.0f;
            float d  = cv - pm;
            ss += d * d;
            pc += pm;
        }
        const int o = (chunk * 2 + half) * (B_ * C_) + b * C_ + c;
        ssum_out[o] = ss;
        pcnt_out[o] = pc;
    }
}

// ---------------------------------------------------------------------------
// Single-block finalize.  Deterministic fixed-order tree reductions only.
// ---------------------------------------------------------------------------
__device__ float block_reduce_256(float v, float* red) {
    const int tid = threadIdx.x;
    __syncthreads();          // protect reuse of `red`
    red[tid] = v;
    __syncthreads();
    for (int s = 128; s > 0; s >>= 1) {
        if (tid < s) red[tid] += red[tid + s];
        __syncthreads();
    }
    return red[0];
}

__global__ void finalize(const float* __restrict__ score_act,
                         const float* __restrict__ score_bkg,
                         const float* __restrict__ label,
                         const float* __restrict__ feat_part /*2048*/,
                         const float* __restrict__ ssum /*[16][6400]*/,
                         const float* __restrict__ pcnt /*[16][6400]*/,
                         float* __restrict__ out) {
    __shared__ float red[256];
    __shared__ float rowsum[B_];
    __shared__ float norms[NBT];
    const int tid = threadIdx.x;

    // label row sums
    if (tid < B_) {
        float s = 0.0f;
        const float* lr = label + tid * C_;
        for (int c = 0; c < C_; ++c) s += lr[c];
        rowsum[tid] = s;
    }
    __syncthreads();

    // BCE losses (mean over 6400 elements; log clamped at -100)
    float acc_cls = 0.0f, acc_be = 0.0f;
    for (int i = tid; i < B_ * C_; i += 256) {
        const int b = i / C_;
        float y  = label[i] / rowsum[b];
        float p  = score_act[i];
        float lp  = fmaxf(logf(p), -100.0f);
        float l1p = fmaxf(logf(1.0f - p), -100.0f);
        acc_cls -= y * lp + (1.0f - y) * l1p;

        float q  = score_bkg[i];
        float lq  = fmaxf(logf(q), -100.0f);
        float l1q = fmaxf(logf(1.0f - q), -100.0f);
        const float yb = 1.0f / (float)C_;
        acc_be -= yb * lq + (1.0f - yb) * l1q;
    }
    float loss_cls = block_reduce_256(acc_cls, red) / (float)(B_ * C_);
    float loss_be  = block_reduce_256(acc_be,  red) / (float)(B_ * C_);

    // feature norms from per-block partials of ||mean||^2
    if (tid < NBT) {
        float s = 0.0f;
        #pragma unroll
        for (int j = 0; j < CHUNKS_PER_BT; ++j) s += feat_part[tid * CHUNKS_PER_BT + j];
        norms[tid] = sqrtf(s);
    }
    __syncthreads();

    float um = 0.0f;
    if (tid < B_) {
        float la = fmaxf(100.0f - norms[tid], 0.0f);   // act margin
        float lb = norms[B_ + tid];                    // bkg magnitude
        float v  = la + lb;
        um = v * v;
    }
    float loss_um = block_reduce_256(um, red) / (float)B_;

    // sup: sum 16 chunk partials per (b,c), then sqrt / valid mask
    float accS = 0.0f, accC = 0.0f;
    for (int i = tid; i < B_ * C_; i += 256) {
        float ss = 0.0f, pc = 0.0f;
        #pragma unroll
        for (int ch = 0; ch < SUP_SLOTS; ++ch) {
            ss += ssum[ch * (B_ * C_) + i];
            pc += pcnt[ch * (B_ * C_) + i];
        }
        if (pc > 0.0f) { accS += sqrtf(ss); accC += 1.0f; }
    }
    float supS = block_reduce_256(accS, red);
    float supC = block_reduce_256(accC, red);
    float loss_sup = supS / supC;

    if (tid == 0)
        out[0] = loss_cls + 0.0005f * loss_um + 0.2f * loss_be + 1.0f * loss_sup;
}

// ---------------------------------------------------------------------------
extern "C" void kernel_launch(void* const* d_in, const int* in_sizes, int n_in,
                              void* d_out, int out_size, void* d_ws, size_t ws_size,
                              hipStream_t stream) {
    const float* score_act = (const float*)d_in[0];
    const float* score_bkg = (const float*)d_in[1];
    const float* feat_act  = (const float*)d_in[2];
    const float* feat_bkg  = (const float*)d_in[3];
    const float* label     = (const float*)d_in[4];
    const float* gt        = (const float*)d_in[5];
    const float* cas       = (const float*)d_in[6];
    float* out = (float*)d_out;

    float* ws      = (float*)d_ws;
    float* ws_feat = ws;                                // 2048 floats
    float* ws_ssum = ws + 2048;                         // 16*6400 = 102400
    float* ws_pcnt = ws_ssum + SUP_SLOTS * B_ * C_;     // 102400

    fused_partials<<<FEAT_BLOCKS + SUP_BLOCKS, 256, 0, stream>>>(
        feat_act, feat_bkg, gt, cas, ws_feat, ws_ssum, ws_pcnt);
    finalize<<<1, 256, 0, stream>>>(score_act, score_bkg, label,
                                    ws_feat, ws_ssum, ws_pcnt, out);
}